// SparseAttention_42236708388940
// MI455X (gfx1250) — compile-verified
//
#include <hip/hip_runtime.h>

#define IN_DIM 256
#define HID    128
#define XS_LD  260   // padded LDS leading dim (260 % 64 != 0 -> no bank conflicts)

typedef float v2f __attribute__((ext_vector_type(2)));
typedef float v8f __attribute__((ext_vector_type(8)));

// ---------------------------------------------------------------- zero init
__global__ void SA_zero_kernel(float* __restrict__ out, float* __restrict__ rowsum,
                               int out_n, int n) {
    int i = blockIdx.x * blockDim.x + threadIdx.x;
    if (i < out_n)            out[i] = 0.0f;
    else if (i < out_n + n)   rowsum[i - out_n] = 0.0f;
}

// ---------------------------------------------------------------- xh = x @ W  (WMMA f32 16x16x4)
// grid.x = n/16 blocks, 256 threads = 8 waves; wave w computes cols [16w, 16w+16)
__global__ __launch_bounds__(256) void SA_gemm_kernel(const float* __restrict__ x,
                                                      const float* __restrict__ W,
                                                      float* __restrict__ xh) {
    __shared__ float xs[16 * XS_LD];
    const int rowbase = blockIdx.x * 16;
    const int tid = threadIdx.x;

    // cooperative load: 16 rows x 256 cols of x -> LDS, 16 floats (4x float4) per thread
    {
        const int r  = tid >> 4;          // 0..15
        const int c0 = (tid & 15) << 4;   // 0,16,...,240
        const float4* src = (const float4*)(x + (size_t)(rowbase + r) * IN_DIM + c0);
        float4* dst = (float4*)(xs + r * XS_LD + c0);
        dst[0] = src[0]; dst[1] = src[1]; dst[2] = src[2]; dst[3] = src[3];
    }
    __syncthreads();

    const int lane = tid & 31;
    const int n0   = (tid >> 5) << 4;     // column tile base for this wave
    const int m    = lane & 15;           // row (A) / col (B,D) within tile
    const int half = lane >> 4;           // 0: K=0,1  1: K=2,3 (A-frag layout, ISA 7.12.2)
    const int koff = half << 1;

    v8f c = {0.f, 0.f, 0.f, 0.f, 0.f, 0.f, 0.f, 0.f};

    #pragma unroll 4
    for (int k = 0; k < IN_DIM; k += 4) {
        v2f a, b;
        // A 16x4 f32: lane m (half 0) -> A[m][k], A[m][k+1]; half 1 -> A[m][k+2], A[m][k+3]
        a.x = xs[m * XS_LD + k + koff];
        a.y = xs[m * XS_LD + k + koff + 1];
        // B 4x16 f32: lane m (half 0) -> B[k][n0+m], B[k+1][n0+m]; half 1 -> rows k+2,k+3
        b.x = W[(size_t)(k + koff)     * HID + n0 + m];
        b.y = W[(size_t)(k + koff + 1) * HID + n0 + m];
        c = __builtin_amdgcn_wmma_f32_16x16x4_f32(false, a, false, b, (short)0, c,
                                                  false, false);
    }

    // D 16x16 f32: vgpr r, lanes 0-15 -> M=r, lanes 16-31 -> M=r+8, N = lane&15
    const int row0 = rowbase + (half << 3);
    #pragma unroll
    for (int r = 0; r < 8; ++r)
        xh[(size_t)(row0 + r) * HID + n0 + m] = c[r];
}

// ---------------------------------------------------------------- s1[i]=xh[i].a1, s2[i]=xh[i].a2
__global__ void SA_dots_kernel(const float* __restrict__ xh, const float* __restrict__ a,
                               float* __restrict__ s1, float* __restrict__ s2, int n) {
    int i = blockIdx.x * blockDim.x + threadIdx.x;
    if (i >= n) return;
    const float4* h4 = (const float4*)(xh + (size_t)i * HID);
    const float4* a1 = (const float4*)(a);
    const float4* a2 = (const float4*)(a + HID);
    float acc1 = 0.f, acc2 = 0.f;
    #pragma unroll 8
    for (int j = 0; j < HID / 4; ++j) {
        float4 h = h4[j], u = a1[j], v = a2[j];
        acc1 += h.x * u.x + h.y * u.y + h.z * u.z + h.w * u.w;
        acc2 += h.x * v.x + h.y * v.y + h.z * v.z + h.w * v.w;
    }
    s1[i] = acc1;
    s2[i] = acc2;
}

// ---------------------------------------------------------------- edge pass 1: e, rowsum
__global__ void SA_edge1_kernel(const long long* __restrict__ edges,
                                const float* __restrict__ s1, const float* __restrict__ s2,
                                float* __restrict__ ee, float* __restrict__ rowsum, int E) {
    int e = blockIdx.x * blockDim.x + threadIdx.x;
    if (e >= E) return;
    int src = (int)edges[e];
    int dst = (int)edges[(size_t)E + e];
    float s = s1[src] + s2[dst];
    float l = (s >= 0.f) ? s : 0.01f * s;     // jax.nn.leaky_relu default slope
    float v = expf(-l);
    ee[e] = v;
    unsafeAtomicAdd(&rowsum[src], v);         // native global_atomic_add_f32
}

// ---------------------------------------------------------------- edge pass 2: out scatter
// one wave32 per edge; lane handles 4 columns (float4 gather + 4 f32 atomic adds)
__global__ __launch_bounds__(256) void SA_edge2_kernel(const long long* __restrict__ edges,
                                                       const float* __restrict__ xh,
                                                       const float* __restrict__ ee,
                                                       const float* __restrict__ rowsum,
                                                       float* __restrict__ out, int E) {
    int gid  = blockIdx.x * blockDim.x + threadIdx.x;
    int e    = gid >> 5;
    int lane = threadIdx.x & 31;
    if (e >= E) return;
    int src = (int)edges[e];
    int dst = (int)edges[(size_t)E + e];
    float w = ee[e] / (rowsum[dst] + 1e-5f);
    float4 h = ((const float4*)(xh + (size_t)dst * HID))[lane];
    float* o = out + (size_t)src * HID + (lane << 2);
    unsafeAtomicAdd(o + 0, w * h.x);
    unsafeAtomicAdd(o + 1, w * h.y);
    unsafeAtomicAdd(o + 2, w * h.z);
    unsafeAtomicAdd(o + 3, w * h.w);
}

// ---------------------------------------------------------------- launch
extern "C" void kernel_launch(void* const* d_in, const int* in_sizes, int n_in,
                              void* d_out, int out_size, void* d_ws, size_t ws_size,
                              hipStream_t stream) {
    const float*      x     = (const float*)d_in[0];
    const long long*  edges = (const long long*)d_in[1];   // (2,E) int64 per reference
    const float*      W     = (const float*)d_in[2];
    const float*      a     = (const float*)d_in[3];       // (1, 2*HID)
    float*            out   = (float*)d_out;

    const int n = in_sizes[0] / IN_DIM;   // 20000
    const int E = in_sizes[1] / 2;        // 640000

    // workspace layout (floats): xh | s1 | s2 | rowsum | ee   (~13 MB)
    float* ws     = (float*)d_ws;
    float* xh     = ws;
    float* s1     = xh + (size_t)n * HID;
    float* s2     = s1 + n;
    float* rowsum = s2 + n;
    float* ee     = rowsum + n;

    const int ztotal = out_size + n;
    SA_zero_kernel<<<(ztotal + 255) / 256, 256, 0, stream>>>(out, rowsum, out_size, n);

    SA_gemm_kernel<<<n / 16, 256, 0, stream>>>(x, W, xh);     // n % 16 == 0 (20000)

    SA_dots_kernel<<<(n + 255) / 256, 256, 0, stream>>>(xh, a, s1, s2, n);

    SA_edge1_kernel<<<(E + 255) / 256, 256, 0, stream>>>(edges, s1, s2, ee, rowsum, E);

    // one wave per edge -> E*32 threads
    SA_edge2_kernel<<<(E * 32 + 255) / 256, 256, 0, stream>>>(edges, xh, ee, rowsum, out, E);
}